// SoftCrossEntropyLoss_65403761983716
// MI455X (gfx1250) — compile-verified
//
#include <hip/hip_runtime.h>

typedef float v2f __attribute__((ext_vector_type(2)));
typedef float v8f __attribute__((ext_vector_type(8)));

#define NBLK 1024
#define TPB  256

__device__ __forceinline__ float max4(float4 v) {
    return fmaxf(fmaxf(v.x, v.y), fmaxf(v.z, v.w));
}

__global__ void __launch_bounds__(TPB)
sce_partial_kernel(const float* __restrict__ x, const float* __restrict__ t,
                   float* __restrict__ ws, int n) {
    const int lane = threadIdx.x & 31;
    const int wave = threadIdx.x >> 5;
    const int gid  = blockIdx.x * TPB + threadIdx.x;
    const int gstride = gridDim.x * TPB;

    float acc = 0.0f;

    // Each lane owns whole rows (128 B = 8 x b128 loads). Consecutive lanes ->
    // consecutive rows, so a wave streams a contiguous 4 KB window per operand.
    for (int r = gid; r < n; r += gstride) {
        const float4* __restrict__ xr = reinterpret_cast<const float4*>(x) + (size_t)r * 8;
        const float4* __restrict__ tr = reinterpret_cast<const float4*>(t) + (size_t)r * 8;

        float4 a[8];
#pragma unroll
        for (int i = 0; i < 8; ++i) a[i] = xr[i];          // global_load_b128 x8

        float m = max4(a[0]);
#pragma unroll
        for (int i = 1; i < 8; ++i) m = fmaxf(m, max4(a[i]));

        float s = 0.0f;
#pragma unroll
        for (int i = 0; i < 8; ++i) {
            s += __expf(a[i].x - m);
            s += __expf(a[i].y - m);
            s += __expf(a[i].z - m);
            s += __expf(a[i].w - m);
        }

        // loss = T*(m + log(sum_exp)) - dot(t, x), with T = sum(t)
        float T = 0.0f, dot = 0.0f;
#pragma unroll
        for (int i = 0; i < 8; ++i) {
            float4 tv = tr[i];                             // global_load_b128 x8
            T   += (tv.x + tv.y) + (tv.z + tv.w);
            dot += tv.x * a[i].x + tv.y * a[i].y + tv.z * a[i].z + tv.w * a[i].w;
        }
        acc += T * (m + __logf(s)) - dot;
    }

    // ---- Intra-wave 32->1 reduction via V_WMMA_F32_16X16X4_F32 (fp32, lossless)
    // A (16x4 f32, 2 VGPRs): VGPR0 lanes0-15 = A[m,0]=acc[m]; lanes16-31 = A[m,2]=acc[m+16].
    // VGPR1 (K=1/K=3) = 0.  B = ones(4x16)  =>  D[m,n] = acc[m] + acc[m+16] for all n.
    v2f A; A[0] = acc;  A[1] = 0.0f;
    v2f B; B[0] = 1.0f; B[1] = 1.0f;
    v8f C = {};
    v8f D = __builtin_amdgcn_wmma_f32_16x16x4_f32(
        /*neg_a=*/false, A, /*neg_b=*/false, B,
        /*c_mod=*/(short)0, C, /*reuse_a=*/false, /*reuse_b=*/false);

    // Lane l<16, VGPR r holds D[r,l] = acc[r]+acc[r+16]; lane>=16 holds rows 8..15.
    float p = ((D[0] + D[1]) + (D[2] + D[3])) + ((D[4] + D[5]) + (D[6] + D[7]));
    p += __shfl_xor(p, 16, 32);     // combine the two half-wave row groups

    __shared__ float wsum[TPB / 32];
    if (lane == 0) wsum[wave] = p;
    __syncthreads();
    if (threadIdx.x == 0) {
        float b = 0.0f;
#pragma unroll
        for (int i = 0; i < TPB / 32; ++i) b += wsum[i];
        ws[blockIdx.x] = b;
    }
}

__global__ void __launch_bounds__(TPB)
sce_finish_kernel(const float* __restrict__ ws, int nblk, float* __restrict__ out,
                  float inv_n) {
    __shared__ float sm[TPB];
    float s = 0.0f;
    for (int i = threadIdx.x; i < nblk; i += TPB) s += ws[i];
    sm[threadIdx.x] = s;
    __syncthreads();
#pragma unroll
    for (int off = TPB / 2; off > 0; off >>= 1) {
        if (threadIdx.x < off) sm[threadIdx.x] += sm[threadIdx.x + off];
        __syncthreads();
    }
    if (threadIdx.x == 0) out[0] = sm[0] * inv_n;
}

extern "C" void kernel_launch(void* const* d_in, const int* in_sizes, int n_in,
                              void* d_out, int out_size, void* d_ws, size_t ws_size,
                              hipStream_t stream) {
    const float* x = (const float*)d_in[0];
    const float* t = (const float*)d_in[1];
    float* out = (float*)d_out;
    float* ws  = (float*)d_ws;

    const int n = in_sizes[0] / 32;   // rows (K = 32 fixed)

    sce_partial_kernel<<<NBLK, TPB, 0, stream>>>(x, t, ws, n);
    sce_finish_kernel<<<1, TPB, 0, stream>>>(ws, NBLK, out, 1.0f / (float)n);
}